// StatePredictor_36086315221639
// MI455X (gfx1250) — compile-verified
//
#include <hip/hip_runtime.h>

// ---------------------------------------------------------------------------
// StatePredictor for MI455X (gfx1250, wave32).
//   Phase 1: brute-force KNN top-11 per point, candidate range split into 4
//            segments per query for 4x wave-level parallelism (2048 waves),
//            register bubble-insert + LDS merge of sorted segment lists.
//   Phase 2: feature gather + 3-layer f32 MLP via V_WMMA_F32_16X16X4_F32.
//            One wave = 16 points.  K=66 padded to 68 (17 chunks of 4).
// ---------------------------------------------------------------------------

#define BATCH   4
#define NPTS    4096
#define KNN     10
#define KP1     11
#define LEN_IN  66
#define LEN_PAD 68
#define TOTAL   (BATCH * NPTS)

#define SEG     4                 // candidate segments per query
#define QPB     64                // queries per block
#define SEGLEN  (NPTS / SEG)      // 1024 candidates per segment
#define CTILE   64                // candidate tile per segment group

typedef __attribute__((ext_vector_type(2))) float v2f;
typedef __attribute__((ext_vector_type(8))) float v8f;

__device__ __forceinline__ v8f wmma_f32_16x16x4(v2f a, v2f b, v8f c) {
    // (neg_a, A, neg_b, B, c_mod, C, reuse_a, reuse_b)
    return __builtin_amdgcn_wmma_f32_16x16x4_f32(false, a, false, b,
                                                 (short)0, c, false, false);
}

// ---------------------------------------------------------------------------
// Kernel 1: per-point top-(K+1) nearest neighbors (ascending d2, self first).
// d2 = sq_i + sq_j - 2*dot  -- identical formula to the reference so the
// ordering (incl. exact-zero self distance) matches.
// Block = 256 threads = 4 segment-groups x 64 queries.
// ---------------------------------------------------------------------------
__global__ __launch_bounds__(256)
void knn_topk_kernel(const float* __restrict__ pos, int* __restrict__ nn_out) {
    __shared__ float4 sP[SEG][CTILE];      // (x, y, z, |p|^2) per candidate
    __shared__ float  sD[SEG][QPB][KP1];   // per-segment sorted distances
    __shared__ int    sI[SEG][QPB][KP1];   // per-segment sorted indices

    const int tid = threadIdx.x;
    const int s   = tid >> 6;              // segment group 0..3
    const int q   = tid & (QPB - 1);       // query slot within block
    const int point = blockIdx.x * QPB + q;     // block fully inside one batch
    const int b   = point >> 12;
    const int i   = point & (NPTS - 1);
    const float* pb = pos + (size_t)b * NPTS * 3;

    const float px = pb[i * 3 + 0];
    const float py = pb[i * 3 + 1];
    const float pz = pb[i * 3 + 2];
    const float sqi = px * px + py * py + pz * pz;

    float bd[KP1];
    int   bi[KP1];
#pragma unroll
    for (int t = 0; t < KP1; ++t) { bd[t] = 3.4028235e38f; bi[t] = 0; }

    const int segBase = s * SEGLEN;
    for (int jt = 0; jt < SEGLEN; jt += CTILE) {
        const int j   = segBase + jt + q;  // 64 threads of group s load a tile
        const float x = pb[j * 3 + 0];
        const float y = pb[j * 3 + 1];
        const float z = pb[j * 3 + 2];
        __syncthreads();                   // protect previous tile
        sP[s][q] = make_float4(x, y, z, x * x + y * y + z * z);
        __syncthreads();

#pragma unroll 4
        for (int jj = 0; jj < CTILE; ++jj) {
            const float4 qc = sP[s][jj];   // ds_load_b128, wave-broadcast
            const float d2 = sqi + qc.w - 2.0f * (px * qc.x + py * qc.y + pz * qc.z);
            if (d2 < bd[KNN]) {            // strict: keep earlier index on tie
                bd[KNN] = d2; bi[KNN] = segBase + jt + jj;
#pragma unroll
                for (int p = KNN; p > 0; --p) {   // one bubble pass re-sorts
                    if (bd[p] < bd[p - 1]) {      // strict: stable on ties
                        float td = bd[p]; bd[p] = bd[p - 1]; bd[p - 1] = td;
                        int   ti = bi[p]; bi[p] = bi[p - 1]; bi[p - 1] = ti;
                    }
                }
            }
        }
    }

    // dump per-segment sorted lists to LDS
#pragma unroll
    for (int t = 0; t < KP1; ++t) { sD[s][q][t] = bd[t]; sI[s][q][t] = bi[t]; }
    __syncthreads();

    // one thread per query merges the 4 sorted lists (index-ascending segment
    // order + strict compares == top_k's lowest-index-first tie-break)
    if (tid < QPB) {
#pragma unroll
        for (int t = 0; t < KP1; ++t) { bd[t] = 3.4028235e38f; bi[t] = 0; }
        for (int ss = 0; ss < SEG; ++ss) {
            for (int t = 0; t < KP1; ++t) {
                const float d = sD[ss][tid][t];
                if (d >= bd[KNN]) break;   // segment list ascending -> done
                bd[KNN] = d; bi[KNN] = sI[ss][tid][t];
#pragma unroll
                for (int p = KNN; p > 0; --p) {
                    if (bd[p] < bd[p - 1]) {
                        float td = bd[p]; bd[p] = bd[p - 1]; bd[p - 1] = td;
                        int   ti = bi[p]; bi[p] = bi[p - 1]; bi[p - 1] = ti;
                    }
                }
            }
        }
        const int pt = blockIdx.x * QPB + tid;
#pragma unroll
        for (int t = 0; t < KP1; ++t) nn_out[(size_t)pt * KP1 + t] = bi[t];
    }
}

// ---------------------------------------------------------------------------
// Kernel 2: gather 66-d features, run 66->32->16->6 MLP with f32 WMMA.
// Block = 128 threads = 4 waves; each wave owns a 16-point tile.
// ---------------------------------------------------------------------------
__global__ __launch_bounds__(128)
void mlp_wmma_kernel(const float* __restrict__ pos,
                     const float* __restrict__ vel,
                     const float* __restrict__ init_cfg,
                     const float* __restrict__ W1, const float* __restrict__ b1,
                     const float* __restrict__ W2, const float* __restrict__ b2,
                     const float* __restrict__ W3, const float* __restrict__ b3,
                     const int*   __restrict__ nn_idx,
                     float*       __restrict__ out) {
    __shared__ float sW1[32][LEN_PAD];       // W1[ch][feat], K-padded with zeros
    __shared__ float sb1[32];
    __shared__ float sW2[16][32];
    __shared__ float sb2[16];
    __shared__ float sW3[16][16];            // out channels padded 6 -> 16
    __shared__ float sb3[16];
    __shared__ float sFeat[4][16][LEN_PAD];  // per-wave 16 x 68 feature tile
    __shared__ float sH1[4][16][32];
    __shared__ float sH2[4][16][16];

    const int tid  = threadIdx.x;
    const int wave = tid >> 5;
    const int lane = tid & 31;

    // ---- stage weights into LDS (cooperative, zero-padded) ----
    for (int idx = tid; idx < 32 * LEN_PAD; idx += 128) {
        const int r = idx / LEN_PAD, c = idx % LEN_PAD;
        sW1[r][c] = (c < LEN_IN) ? W1[r * LEN_IN + c] : 0.0f;
    }
    for (int idx = tid; idx < 16 * 32; idx += 128) sW2[idx >> 5][idx & 31] = W2[idx];
    for (int idx = tid; idx < 16 * 16; idx += 128) {
        const int r = idx >> 4, c = idx & 15;
        sW3[r][c] = (r < 6) ? W3[r * 16 + c] : 0.0f;
    }
    if (tid < 32) sb1[tid] = b1[tid];
    if (tid < 16) sb2[tid] = b2[tid];
    if (tid < 16) sb3[tid] = (tid < 6) ? b3[tid] : 0.0f;

    // ---- gather features: lane m handles point m; half 0 -> pos_off+init,
    //      half 1 -> velocities ----
    const int m        = lane & 15;
    const int half     = lane >> 4;
    const int tileBase = blockIdx.x * 64 + wave * 16;
    const int point    = tileBase + m;
    const int b        = point >> 12;
    const int i        = point & (NPTS - 1);
    const float* pb    = pos + (size_t)b * NPTS * 3;
    const float* vb    = vel + (size_t)b * NPTS * 3;
    const int*   nn    = nn_idx + (size_t)point * KP1;
    float* frow        = &sFeat[wave][m][0];

    if (half == 0) {
        const float px = pb[i * 3 + 0];
        const float py = pb[i * 3 + 1];
        const float pz = pb[i * 3 + 2];
#pragma unroll
        for (int k = 0; k < KNN; ++k) {          // skip self (slot 0)
            const int j = nn[k + 1];
            frow[3 * k + 0] = pb[j * 3 + 0] - px;
            frow[3 * k + 1] = pb[j * 3 + 1] - py;
            frow[3 * k + 2] = pb[j * 3 + 2] - pz;
        }
        frow[63] = init_cfg[b * 3 + 0];
        frow[64] = init_cfg[b * 3 + 1];
        frow[65] = init_cfg[b * 3 + 2];
        frow[66] = 0.0f;                          // K padding 66 -> 68
        frow[67] = 0.0f;
    } else {
#pragma unroll
        for (int k = 0; k < KP1; ++k) {          // includes self (slot 0)
            const int j = nn[k];
            frow[30 + 3 * k + 0] = vb[j * 3 + 0];
            frow[30 + 3 * k + 1] = vb[j * 3 + 1];
            frow[30 + 3 * k + 2] = vb[j * 3 + 2];
        }
    }
    __syncthreads();

    // WMMA fragment indexing (per ISA 16x16x4 f32 layouts):
    //   A:  lanes 0-15 hold K={0,1}, lanes 16-31 hold K={2,3}  -> koff
    //   B:  column n = lane%16; same K split as A
    //   C/D: column n = lane%16; VGPR r = row (r | r+8 per half) -> rbase
    const int n     = m;
    const int koff  = half * 2;
    const int rbase = half * 8;

    // ---- layer 1: [16x68] x [68x32] ----
    v8f c0 = {};
    v8f c1 = {};
#pragma unroll
    for (int kc = 0; kc < LEN_PAD / 4; ++kc) {
        const int kb = kc * 4 + koff;
        const v2f a  = *(const v2f*)&sFeat[wave][m][kb];
        const v2f bA = *(const v2f*)&sW1[n][kb];        // channels 0-15
        const v2f bB = *(const v2f*)&sW1[n + 16][kb];   // channels 16-31
        c0 = wmma_f32_16x16x4(a, bA, c0);
        c1 = wmma_f32_16x16x4(a, bB, c1);
    }
    {
        const float bias0 = sb1[n], bias1 = sb1[n + 16];
#pragma unroll
        for (int r = 0; r < 8; ++r) {
            sH1[wave][rbase + r][n]      = c0[r] + bias0;
            sH1[wave][rbase + r][n + 16] = c1[r] + bias1;
        }
    }
    __syncthreads();

    // ---- layer 2: [16x32] x [32x16] ----
    v8f c2 = {};
#pragma unroll
    for (int kc = 0; kc < 8; ++kc) {
        const int kb = kc * 4 + koff;
        const v2f a  = *(const v2f*)&sH1[wave][m][kb];
        const v2f bb = *(const v2f*)&sW2[n][kb];
        c2 = wmma_f32_16x16x4(a, bb, c2);
    }
    {
        const float bias = sb2[n];
#pragma unroll
        for (int r = 0; r < 8; ++r) sH2[wave][rbase + r][n] = c2[r] + bias;
    }
    __syncthreads();

    // ---- layer 3: [16x16] x [16x6(pad 16)] ----
    v8f c3 = {};
#pragma unroll
    for (int kc = 0; kc < 4; ++kc) {
        const int kb = kc * 4 + koff;
        const v2f a  = *(const v2f*)&sH2[wave][m][kb];
        const v2f bb = *(const v2f*)&sW3[n][kb];
        c3 = wmma_f32_16x16x4(a, bb, c3);
    }

    // ---- epilogue: +bias, +position on channels 0..2, store [B,N,6] ----
    if (n < 6) {
        const float bias = sb3[n];
#pragma unroll
        for (int r = 0; r < 8; ++r) {
            const int pt = tileBase + rbase + r;
            float val = c3[r] + bias;
            if (n < 3) val += pos[(size_t)pt * 3 + n];
            out[(size_t)pt * 6 + n] = val;
        }
    }
}

// ---------------------------------------------------------------------------
extern "C" void kernel_launch(void* const* d_in, const int* in_sizes, int n_in,
                              void* d_out, int out_size, void* d_ws, size_t ws_size,
                              hipStream_t stream) {
    (void)in_sizes; (void)n_in; (void)out_size; (void)ws_size;

    const float* pos  = (const float*)d_in[0];
    const float* vel  = (const float*)d_in[1];
    const float* icfg = (const float*)d_in[2];
    const float* W1   = (const float*)d_in[3];
    const float* b1   = (const float*)d_in[4];
    const float* W2   = (const float*)d_in[5];
    const float* b2   = (const float*)d_in[6];
    const float* W3   = (const float*)d_in[7];
    const float* b3   = (const float*)d_in[8];
    float* out        = (float*)d_out;
    int*   nn         = (int*)d_ws;   // B*N*11 ints = 720 KB scratch

    knn_topk_kernel<<<TOTAL / QPB, 256, 0, stream>>>(pos, nn);
    mlp_wmma_kernel<<<TOTAL / 64, 128, 0, stream>>>(pos, vel, icfg,
                                                    W1, b1, W2, b2, W3, b3,
                                                    nn, out);
}